// Community_Pool_72146860638840
// MI455X (gfx1250) — compile-verified
//
#include <hip/hip_runtime.h>
#include <math.h>

typedef float v2f __attribute__((ext_vector_type(2)));
typedef float v8f __attribute__((ext_vector_type(8)));

#define CD    64      // channels
#define NPG   1000    // nodes per graph
#define KTOP  500     // top-k
#define SORTN 1024    // padded sort size

// ---------------------------------------------------------------------------
// h = x @ W   via v_wmma_f32_16x16x4_f32 (fp32 WMMA, wave32)
// Block: 128 threads = 4 waves. Block computes a 16-row slab of h; wave w
// computes columns [16w, 16w+16). W (64x64, 16 KB) staged in LDS.
// A 16x4 layout: lanes0-15 hold K={k0,k0+1}, lanes16-31 hold K={k0+2,k0+3}.
// C/D 16x16:      vgpr v, lanes0-15 -> M=v, lanes16-31 -> M=v+8, N=lane&15.
// ---------------------------------------------------------------------------
__global__ __launch_bounds__(128) void k_gemm(const float* __restrict__ x,
                                              const float* __restrict__ Wm,
                                              float* __restrict__ h, int N) {
  __shared__ float sW[CD * CD];
  const int tid = threadIdx.x;
  for (int i = tid; i < CD * CD; i += 128) sW[i] = Wm[i];
  __syncthreads();

  const int wv   = tid >> 5;           // wave id: column tile
  const int lane = tid & 31;
  const int row0 = blockIdx.x << 4;    // 16 rows per block
  const int m    = lane & 15;
  const int kh   = (lane >> 4) << 1;   // 0 or 2
  const int ncol = (wv << 4) + m;

  const float* __restrict__ xr = x + (size_t)(row0 + m) * CD;
  v8f acc = {};
#pragma unroll
  for (int k0 = 0; k0 < CD; k0 += 4) {
    v2f a, b;
    a.x = xr[k0 + kh];
    a.y = xr[k0 + kh + 1];
    b.x = sW[(k0 + kh) * CD + ncol];
    b.y = sW[(k0 + kh + 1) * CD + ncol];
    acc = __builtin_amdgcn_wmma_f32_16x16x4_f32(false, a, false, b,
                                                (short)0, acc, false, false);
  }
  const int mb = (lane >> 4) << 3;     // 0 or 8
#pragma unroll
  for (int v = 0; v < 8; ++v)
    h[(size_t)(row0 + mb + v) * CD + ncol] = acc[v];
}

// ---------------------------------------------------------------------------
// Degree (with self-loop) -> dinv = rsqrt(deg)
// ---------------------------------------------------------------------------
__global__ void k_deg_init(float* deg, int N) {
  int i = blockIdx.x * blockDim.x + threadIdx.x;
  if (i < N) deg[i] = 1.0f;            // self loop contributes 1
}
__global__ void k_deg_acc(const int* __restrict__ dst, float* deg, int E) {
  int e = blockIdx.x * blockDim.x + threadIdx.x;
  if (e < E) atomicAdd(&deg[dst[e]], 1.0f);
}
__global__ void k_dinv(const float* __restrict__ deg, float* dinv, int N) {
  int i = blockIdx.x * blockDim.x + threadIdx.x;
  if (i < N) dinv[i] = rsqrtf(deg[i]); // deg >= 1 always
}

// out[i][c] = h[i][c]*dinv[i]^2 + b[c]    (self-loop term + bias)
__global__ void k_out_init(const float* __restrict__ h,
                           const float* __restrict__ dinv,
                           const float* __restrict__ bias,
                           float* __restrict__ outf, int total) {
  int i = blockIdx.x * blockDim.x + threadIdx.x;
  if (i < total) {
    int r = i >> 6;
    float dv = dinv[r];
    outf[i] = h[i] * dv * dv + bias[i & 63];
  }
}

// Edge scatter: 16 lanes per edge, 4 channels per lane (float4-granular).
// h and outf both fit in the 192MB L2, so these atomics resolve in L2.
__global__ void k_scatter(const int* __restrict__ src,
                          const int* __restrict__ dst,
                          const float* __restrict__ h,
                          const float* __restrict__ dinv,
                          float* __restrict__ outf, int E) {
  int t = blockIdx.x * blockDim.x + threadIdx.x;
  int e = t >> 4;
  if (e >= E) return;
  int c0 = (t & 15) << 2;
  int s = src[e], d = dst[e];
  float norm = dinv[s] * dinv[d];
  const float* hs = h + (size_t)s * CD + c0;
  float* op = outf + (size_t)d * CD + c0;
#pragma unroll
  for (int c = 0; c < 4; ++c) atomicAdd(op + c, hs[c] * norm);
}

// ---------------------------------------------------------------------------
// Per-graph: centroid -> cosine score -> softmax KL -> bitonic argsort (1024)
// Writes kl, indices_, perm, batch_new; stashes score + int perm for later.
// ---------------------------------------------------------------------------
__global__ __launch_bounds__(256) void k_score_sort(
    const float* __restrict__ outf, float* __restrict__ score_ws,
    int* __restrict__ perm_ws, float* __restrict__ o_batch,
    float* __restrict__ o_perm, float* __restrict__ o_kl,
    float* __restrict__ o_idx) {
  __shared__ float cen[CD];
  __shared__ float psum[256];
  __shared__ float red[256];
  __shared__ float ss[SORTN];
  __shared__ int   si[SORTN];
  __shared__ float sc_cnorm, sc_mx, sc_S;

  const int g = blockIdx.x;
  const int tid = threadIdx.x;
  const float* gx = outf + (size_t)g * NPG * CD;

  // centroid: 4 groups of 64 threads, group strides over nodes
  {
    int c = tid & 63, grp = tid >> 6;
    float s = 0.f;
    for (int n = grp; n < NPG; n += 4) s += gx[(size_t)n * CD + c];
    psum[tid] = s;
  }
  __syncthreads();
  if (tid < CD)
    cen[tid] = (psum[tid] + psum[tid + 64] + psum[tid + 128] + psum[tid + 192])
               * (1.0f / (float)NPG);
  __syncthreads();
  if (tid == 0) {
    float s = 0.f;
    for (int c = 0; c < CD; ++c) s += cen[c] * cen[c];
    sc_cnorm = sqrtf(s);
  }
  __syncthreads();
  const float cn = sc_cnorm;

  // per-node cosine score (pad to 1024 with -inf)
  for (int n = tid; n < SORTN; n += 256) {
    float sc = -INFINITY;
    if (n < NPG) {
      float num = 0.f, sq = 0.f;
      const float* row = gx + (size_t)n * CD;
#pragma unroll 8
      for (int c = 0; c < CD; ++c) { float v = row[c]; num += v * cen[c]; sq += v * v; }
      sc = num / (sqrtf(sq) * cn + 1e-8f);
      score_ws[(size_t)g * NPG + n] = sc;
    }
    ss[n] = sc; si[n] = n;
  }
  __syncthreads();

  // softmax max
  float lmx = -INFINITY;
  for (int n = tid; n < NPG; n += 256) lmx = fmaxf(lmx, ss[n]);
  red[tid] = lmx; __syncthreads();
  for (int s = 128; s > 0; s >>= 1) {
    if (tid < s) red[tid] = fmaxf(red[tid], red[tid + s]);
    __syncthreads();
  }
  if (tid == 0) sc_mx = red[0];
  __syncthreads();
  const float mx = sc_mx;

  // sum(exp) and sum(exp * (s - mx))
  float lS = 0.f, lK = 0.f;
  for (int n = tid; n < NPG; n += 256) {
    float d = ss[n] - mx, e = __expf(d);
    lS += e; lK += e * d;
  }
  red[tid] = lS; __syncthreads();
  for (int s = 128; s > 0; s >>= 1) { if (tid < s) red[tid] += red[tid + s]; __syncthreads(); }
  if (tid == 0) sc_S = red[0];
  __syncthreads();
  red[tid] = lK; __syncthreads();
  for (int s = 128; s > 0; s >>= 1) { if (tid < s) red[tid] += red[tid + s]; __syncthreads(); }
  if (tid == 0) {
    float S = sc_S;
    o_kl[g] = red[0] / S + logf((float)NPG) - logf(S);
  }
  __syncthreads();

  // bitonic sort, descending score, tie -> lower index first (stable argsort)
  for (int ksz = 2; ksz <= SORTN; ksz <<= 1) {
    for (int j = ksz >> 1; j > 0; j >>= 1) {
      for (int i = tid; i < SORTN; i += 256) {
        int p = i ^ j;
        if (p > i) {
          bool desc = ((i & ksz) == 0);
          float a = ss[i], bv = ss[p];
          int ia = si[i], ib = si[p];
          bool aFirst = (a > bv) || (a == bv && ia < ib);
          if (desc ? !aFirst : aFirst) {
            ss[i] = bv; ss[p] = a; si[i] = ib; si[p] = ia;
          }
        }
      }
      __syncthreads();
    }
  }

  for (int n = tid; n < NPG; n += 256)
    o_idx[(size_t)g * NPG + n] = (float)si[n];
  for (int jj = tid; jj < KTOP; jj += 256) {
    int node = si[jj];
    int pv = g * NPG + node;
    perm_ws[g * KTOP + jj] = pv;
    o_perm[g * KTOP + jj]  = (float)pv;
    o_batch[g * KTOP + jj] = (float)g;
  }
}

// ---------------------------------------------------------------------------
// mapping / gather / edge filter
// ---------------------------------------------------------------------------
__global__ void k_map_init(int* mapping, int N) {
  int i = blockIdx.x * blockDim.x + threadIdx.x;
  if (i < N) mapping[i] = -1;
}
__global__ void k_map_set(const int* __restrict__ perm, int* mapping, int M) {
  int j = blockIdx.x * blockDim.x + threadIdx.x;
  if (j < M) mapping[perm[j]] = j;
}
__global__ void k_xnew(const float* __restrict__ outf,
                       const float* __restrict__ score_ws,
                       const int* __restrict__ perm,
                       float* __restrict__ o_x, int total) {
  int t = blockIdx.x * blockDim.x + threadIdx.x;
  if (t >= total) return;
  int j = t >> 6, c = t & 63;
  int node = perm[j];
  o_x[t] = outf[(size_t)node * CD + c] * tanhf(score_ws[node]);
}
__global__ void k_edges(const int* __restrict__ ei,
                        const float* __restrict__ attr,
                        const int* __restrict__ mapping,
                        float* __restrict__ o_ei, float* __restrict__ o_attr,
                        float* __restrict__ o_mask, int E) {
  int e = blockIdx.x * blockDim.x + threadIdx.x;
  if (e >= E) return;
  int s = mapping[ei[e]];        // row 0 (src)
  int d = mapping[ei[E + e]];    // row 1 (dst)
  o_ei[e]     = (float)s;
  o_ei[E + e] = (float)d;
  bool keep = (s >= 0) && (d >= 0);
  o_mask[e] = keep ? 1.0f : 0.0f;
  o_attr[e] = keep ? attr[e] : 0.0f;
}

// ---------------------------------------------------------------------------
extern "C" void kernel_launch(void* const* d_in, const int* in_sizes, int n_in,
                              void* d_out, int out_size, void* d_ws, size_t ws_size,
                              hipStream_t stream) {
  const float* x    = (const float*)d_in[1];
  const int*   ei   = (const int*)d_in[2];     // [2,E]: row0=src, row1=dst
  const float* attr = (const float*)d_in[3];
  const float* Wm   = (const float*)d_in[5];
  const float* bias = (const float*)d_in[6];

  const int N  = in_sizes[1] / CD;   // 256000
  const int E  = in_sizes[3];        // 4096000
  const int Bn = N / NPG;            // 256
  const int M  = Bn * KTOP;          // 128000

  // workspace carve
  char* wsp = (char*)d_ws;
  auto carve = [&](size_t bytes) -> char* {
    char* p = wsp; wsp += (bytes + 255) & ~(size_t)255; return p;
  };
  float* h       = (float*)carve((size_t)N * CD * 4);
  float* outf    = (float*)carve((size_t)N * CD * 4);
  float* deg     = (float*)carve((size_t)N * 4);
  float* dinv    = (float*)carve((size_t)N * 4);
  float* score   = (float*)carve((size_t)N * 4);
  int*   permi   = (int*)  carve((size_t)M * 4);
  int*   mapping = (int*)  carve((size_t)N * 4);

  // output layout (all float, reference return order)
  float* o       = (float*)d_out;
  float* o_x     = o;                          // [M, 64]
  float* o_ei    = o_x    + (size_t)M * CD;    // [2, E]
  float* o_attr  = o_ei   + (size_t)2 * E;     // [E]
  float* o_mask  = o_attr + (size_t)E;         // [E]
  float* o_batch = o_mask + (size_t)E;         // [M]
  float* o_perm  = o_batch + (size_t)M;        // [M]
  float* o_kl    = o_perm + (size_t)M;         // [Bn]
  float* o_idx   = o_kl   + (size_t)Bn;        // [Bn, NPG]

  const int T = 256;
  k_gemm<<<N / 16, 128, 0, stream>>>(x, Wm, h, N);
  k_deg_init<<<(N + T - 1) / T, T, 0, stream>>>(deg, N);
  k_deg_acc<<<(E + T - 1) / T, T, 0, stream>>>(ei + E, deg, E);
  k_dinv<<<(N + T - 1) / T, T, 0, stream>>>(deg, dinv, N);

  int tot = N * CD;
  k_out_init<<<(tot + T - 1) / T, T, 0, stream>>>(h, dinv, bias, outf, tot);
  int st = E * 16;
  k_scatter<<<(st + T - 1) / T, T, 0, stream>>>(ei, ei + E, h, dinv, outf, E);

  k_score_sort<<<Bn, 256, 0, stream>>>(outf, score, permi, o_batch, o_perm,
                                       o_kl, o_idx);

  k_map_init<<<(N + T - 1) / T, T, 0, stream>>>(mapping, N);
  k_map_set<<<(M + T - 1) / T, T, 0, stream>>>(permi, mapping, M);
  k_xnew<<<(M * CD + T - 1) / T, T, 0, stream>>>(outf, score, permi, o_x, M * CD);
  k_edges<<<(E + T - 1) / T, T, 0, stream>>>(ei, attr, mapping, o_ei, o_attr,
                                             o_mask, E);
}